// ConstrainedMeanShiftSelf_52183852647059
// MI455X (gfx1250) — compile-verified
//
#include <hip/hip_runtime.h>
#include <stdint.h>

// Problem constants (from reference setup_inputs)
#define BB   256
#define DD   512
#define KK   128000
#define NPOOL 100000
#define TOPKN 5
#define TOPKP 10

// GEMM/top-k tiling
#define ROWG 4                        // 4 row-groups of 64 rows (=256)
#define NCHUNK 40                     // column chunks over K
#define COLS_PER_CHUNK (KK / NCHUNK)  // 3200
#define CGROUP 64                     // columns processed per iteration (4 N-tiles)
#define NGROUPS (COLS_PER_CHUNK / CGROUP) // 50
#define DCH 64                        // depth chunk (floats) - double buffered
#define NDCH (DD / DCH)               // 8
#define BST 68                        // padded LDS row stride (floats): banks 4n+c, conflict-free

typedef __attribute__((ext_vector_type(2))) float v2f;
typedef __attribute__((ext_vector_type(8))) float v8f;

// Async global->LDS copy (CDNA5 TDM-adjacent path, tracked by ASYNCcnt).
// Moves 16B per lane directly into LDS without a VGPR round-trip.
__device__ __forceinline__ void async_lds_load_b128(float* ldsaddr, const float* gaddr) {
  uint32_t l = (uint32_t)(uintptr_t)ldsaddr;   // low 32 bits = LDS offset
  asm volatile("global_load_async_to_lds_b128 %0, %1, off"
               :: "v"(l), "v"(gaddr)
               : "memory");
}
__device__ __forceinline__ void wait_asynccnt0() {
  asm volatile("s_wait_asynccnt 0" ::: "memory");
}

// ---------------------------------------------------------------------------
// Normalize query -> qn (blocks 0..255) and current_target -> tn (256..511)
__global__ void nrm_kernel(const float* __restrict__ query,
                           const float* __restrict__ ctgt,
                           float* __restrict__ qn, float* __restrict__ tn) {
  __shared__ float red[128];
  const int row = blockIdx.x & 255;
  const bool isT = blockIdx.x >= 256;
  const float* src = (isT ? ctgt : query) + (size_t)row * DD;
  float* dst = (isT ? tn : qn) + (size_t)row * DD;
  float s = 0.f;
  for (int i = threadIdx.x; i < DD; i += 128) { float v = src[i]; s += v * v; }
  red[threadIdx.x] = s;
  __syncthreads();
  for (int w = 64; w > 0; w >>= 1) {
    if (threadIdx.x < w) red[threadIdx.x] += red[threadIdx.x + w];
    __syncthreads();
  }
  const float inv = 1.0f / sqrtf(red[0]);
  for (int i = threadIdx.x; i < DD; i += 128) dst[i] = src[i] * inv;
}

// ---------------------------------------------------------------------------
__global__ void mark_clear(int* __restrict__ mark) {
  int i = blockIdx.x * 256 + threadIdx.x;
  if (i < NPOOL) mark[i] = 0;
}
__global__ void mark_set(const int* __restrict__ indices, int* __restrict__ mark) {
  if (threadIdx.x < BB) mark[indices[threadIdx.x]] = 1;
}
// targets_prime[k] = pool[(pq[n]^mark[n])*N + n], n = k<B ? indices[k] : index_queue[k]
__global__ void tprow_kernel(const int* __restrict__ indices,
                             const int* __restrict__ index_queue,
                             const int* __restrict__ pool_qindex,
                             const int* __restrict__ mark,
                             int* __restrict__ tpRow) {
  int k = blockIdx.x * 256 + threadIdx.x;
  if (k >= KK) return;
  int n = (k < BB) ? indices[k] : index_queue[k];
  int s = pool_qindex[n] ^ mark[n];
  tpRow[k] = s * NPOOL + n;
}

// ---------------------------------------------------------------------------
// Streaming GEMM (dist = 2 - 2*A.Brows) + per-row running top-TK.
// Each wave owns one 16-row M-tile x four 16-col N-tiles (64 cols); one
// A-fragment load feeds 4 independent v_wmma_f32_16x16x4_f32 ops. B tiles
// are staged into double-buffered LDS via global_load_async_to_lds_b128 so
// the async DMA overlaps the WMMA loop.
// MODE 0: A=t_norm, B=queue_eff.  MODE 1: A/B = gathered pool rows (tpRow).
template <int TK, int MODE>
__global__ void gemm_topk_kernel(const float* __restrict__ tn,
                                 const float* __restrict__ queue,
                                 const float* __restrict__ pool,
                                 const int* __restrict__ tpRow,
                                 float* __restrict__ pVal,
                                 int* __restrict__ pIdx) {
  __shared__ float ldsB[2][CGROUP * BST];  // double-buffered 64 rows x 64-depth
  __shared__ float ldsDist[64 * CGROUP];   // 64 A-rows x 64 cols distance block

  const int rg = blockIdx.x / NCHUNK;
  const int chunk = blockIdx.x % NCHUNK;
  const int colBase = chunk * COLS_PER_CHUNK;
  const int tid = threadIdx.x;       // 128 threads = 4 waves (wave32)
  const int wave = tid >> 5;
  const int lane = tid & 31;
  const int half = lane >> 4;        // selects K offset 0 vs 2 in fragments
  const int l16 = lane & 15;

  // A fragment row for this lane (16x4 fp32 layout: lanes 0-15 hold M=0..15
  // K={0,1}; lanes 16-31 hold M=0..15 K={2,3}).
  const int mrow = rg * 64 + wave * 16 + l16;
  const float* Arow = (MODE == 0) ? (tn + (size_t)mrow * DD)
                                  : (pool + (size_t)tpRow[mrow] * DD);

  float bv[TK];
  int bi[TK];
#pragma unroll
  for (int j = 0; j < TK; j++) { bv[j] = 3.0e38f; bi[j] = -1; }

  for (int cg = 0; cg < NGROUPS; cg++) {
    const int c0 = colBase + cg * CGROUP;
    v8f acc[4];
#pragma unroll
    for (int t = 0; t < 4; t++) acc[t] = (v8f){0.f,0.f,0.f,0.f,0.f,0.f,0.f,0.f};

    // Issue async loads for a depth chunk: 64 rows x 64 floats = 1024 x b128,
    // 8 per thread. For fixed i a wave covers 2 rows contiguously.
    auto issue_chunk = [&](int dc) {
      const int d0 = dc * DCH;
      float* buf = ldsB[dc & 1];
#pragma unroll
      for (int i = 0; i < 8; i++) {
        const int idx = i * 128 + tid;
        const int row = idx >> 4;          // 0..63
        const int c4 = idx & 15;
        const int r = c0 + row;
        const float* src;
        if (MODE == 0) src = (r < BB) ? (tn + (size_t)r * DD) : (queue + (size_t)r * DD);
        else           src = pool + (size_t)tpRow[r] * DD;
        async_lds_load_b128(buf + row * BST + c4 * 4, src + d0 + c4 * 4);
      }
    };

    issue_chunk(0);
    wait_asynccnt0();
    __syncthreads();

    for (int dc = 0; dc < NDCH; dc++) {
      if (dc + 1 < NDCH) issue_chunk(dc + 1);  // overlap DMA with WMMA

      const float* buf = ldsB[dc & 1];
      const int d0 = dc * DCH;
      // Depth accumulate: one A-fragment load feeds 4 WMMAs (4 N-tiles)
      for (int kk = 0; kk < DCH; kk += 4) {
        const float2 af = *(const float2*)(Arow + d0 + kk + 2 * half);
        const v2f a = {af.x, af.y};
#pragma unroll
        for (int t = 0; t < 4; t++) {
          const float2 bf = *(const float2*)(buf + (t * 16 + l16) * BST + kk + 2 * half);
          const v2f b = {bf.x, bf.y};
          acc[t] = __builtin_amdgcn_wmma_f32_16x16x4_f32(false, a, false, b,
                                                         (short)0, acc[t],
                                                         false, false);
        }
      }
      wait_asynccnt0();   // next chunk's DMA done (overlapped with loop above)
      __syncthreads();
    }

    // C layout: VGPR r holds (M=r, N=lane) lanes<16, (M=r+8, N=lane-16) else
#pragma unroll
    for (int t = 0; t < 4; t++) {
#pragma unroll
      for (int r = 0; r < 8; r++) {
        const int rl = wave * 16 + r + 8 * half;
        ldsDist[rl * CGROUP + t * 16 + l16] = 2.0f - 2.0f * acc[t][r];
      }
    }
    __syncthreads();

    // Per-row running top-k (thread t<64 owns local row t, scans 64 cols)
    if (tid < 64) {
      for (int j = 0; j < CGROUP; j++) {
        const float v = ldsDist[tid * CGROUP + j];
        if (v < bv[TK - 1]) {
          int p = TK - 1;
          while (p > 0 && bv[p - 1] > v) {
            bv[p] = bv[p - 1]; bi[p] = bi[p - 1]; --p;
          }
          bv[p] = v; bi[p] = c0 + j;
        }
      }
    }
    __syncthreads();   // ldsDist reusable next group
  }

  if (tid < 64) {
    const int grow = rg * 64 + tid;
    float* ov = pVal + ((size_t)grow * NCHUNK + chunk) * TK;
    int*   oi = pIdx + ((size_t)grow * NCHUNK + chunk) * TK;
#pragma unroll
    for (int j = 0; j < TK; j++) { ov[j] = bv[j]; oi[j] = bi[j]; }
  }
}

// ---------------------------------------------------------------------------
template <int TK>
__global__ void merge_kernel(const float* __restrict__ pVal,
                             const int* __restrict__ pIdx,
                             int* __restrict__ outIdx) {
  const int b = threadIdx.x;
  if (b >= BB) return;
  float bv[TK];
  int bi[TK];
#pragma unroll
  for (int j = 0; j < TK; j++) { bv[j] = 3.0e38f; bi[j] = -1; }
  for (int ch = 0; ch < NCHUNK; ch++) {
    const float* v = pVal + ((size_t)b * NCHUNK + ch) * TK;
    const int*  ii = pIdx + ((size_t)b * NCHUNK + ch) * TK;
#pragma unroll
    for (int j = 0; j < TK; j++) {
      const float x = v[j];
      if (x < bv[TK - 1]) {
        int p = TK - 1;
        while (p > 0 && bv[p - 1] > x) {
          bv[p] = bv[p - 1]; bi[p] = bi[p - 1]; --p;
        }
        bv[p] = x; bi[p] = ii[j];
      }
    }
  }
#pragma unroll
  for (int j = 0; j < TK; j++) outIdx[b * TK + j] = bi[j];
}

// ---------------------------------------------------------------------------
// Gather-recompute: dist_q at un_idx, dist_t/dist_q at idx_p; con_idx = 5
// smallest dist_t among idx_p (the -5 scatter makes those strictly minimal
// since dist_t of unit vectors lies in [0,4]).
__global__ void final_kernel(const float* __restrict__ tn,
                             const float* __restrict__ qn,
                             const float* __restrict__ queue,
                             const int* __restrict__ unIdx,
                             const int* __restrict__ ipIdx,
                             const int* __restrict__ labels_bank,
                             const int* __restrict__ labels,
                             float* __restrict__ out) {
  __shared__ float redL[256];
  __shared__ float redP[256];
  const int b = threadIdx.x;
  const float* qrow = qn + (size_t)b * DD;
  const float* trow = tn + (size_t)b * DD;

  float sum_un = 0.f, pur = 0.f;
  for (int j = 0; j < TOPKN; j++) {
    const int k = unIdx[b * TOPKN + j];
    const float* brow = (k < BB) ? tn + (size_t)k * DD : queue + (size_t)k * DD;
    float s = 0.f;
    for (int d = 0; d < DD; d++) s += qrow[d] * brow[d];
    sum_un += 2.f - 2.f * s;
    const int lbl = (k < BB) ? labels[k] : labels_bank[k];
    pur += (lbl == labels[b]) ? 1.f : 0.f;
  }

  float dt[TOPKP], dq[TOPKP];
  for (int j = 0; j < TOPKP; j++) {
    const int k = ipIdx[b * TOPKP + j];
    const float* brow = (k < BB) ? tn + (size_t)k * DD : queue + (size_t)k * DD;
    float st = 0.f, sq = 0.f;
    for (int d = 0; d < DD; d++) { st += trow[d] * brow[d]; sq += qrow[d] * brow[d]; }
    dt[j] = 2.f - 2.f * st;
    dq[j] = 2.f - 2.f * sq;
  }
  float sum_con = 0.f;
  bool used[TOPKP];
#pragma unroll
  for (int j = 0; j < TOPKP; j++) used[j] = false;
  for (int s = 0; s < TOPKN; s++) {
    int am = 0; float mv = 3.0e38f;
    for (int j = 0; j < TOPKP; j++)
      if (!used[j] && dt[j] < mv) { mv = dt[j]; am = j; }
    used[am] = true;
    sum_con += dq[am];
  }

  redL[b] = (sum_con * (1.f / TOPKN) + sum_un * (1.f / TOPKN)) * 0.5f;
  redP[b] = pur * (1.f / TOPKN);
  __syncthreads();
  for (int w = 128; w > 0; w >>= 1) {
    if (b < w) { redL[b] += redL[b + w]; redP[b] += redP[b + w]; }
    __syncthreads();
  }
  if (b == 0) { out[0] = redL[0] * (1.f / BB); out[1] = redP[0] * (1.f / BB); }
}

// ---------------------------------------------------------------------------
static inline char* ws_take(char** p, size_t bytes) {
  char* r = *p;
  *p += (bytes + 255) & ~(size_t)255;
  return r;
}

extern "C" void kernel_launch(void* const* d_in, const int* in_sizes, int n_in,
                              void* d_out, int out_size, void* d_ws, size_t ws_size,
                              hipStream_t stream) {
  const float* query       = (const float*)d_in[0];
  const float* ctgt        = (const float*)d_in[1];
  const float* queue       = (const float*)d_in[2];
  const float* pool        = (const float*)d_in[3];
  const int*   labels_bank = (const int*)d_in[4];
  const int*   index_queue = (const int*)d_in[5];
  const int*   pool_qindex = (const int*)d_in[6];
  const int*   labels      = (const int*)d_in[7];
  const int*   indices     = (const int*)d_in[8];
  float* out = (float*)d_out;

  char* p = (char*)d_ws;
  float* tn    = (float*)ws_take(&p, (size_t)BB * DD * 4);
  float* qn    = (float*)ws_take(&p, (size_t)BB * DD * 4);
  int*   mark  = (int*)ws_take(&p, (size_t)NPOOL * 4);
  int*   tpRow = (int*)ws_take(&p, (size_t)KK * 4);
  float* p0v   = (float*)ws_take(&p, (size_t)BB * NCHUNK * TOPKN * 4);
  int*   p0i   = (int*)ws_take(&p, (size_t)BB * NCHUNK * TOPKN * 4);
  float* p1v   = (float*)ws_take(&p, (size_t)BB * NCHUNK * TOPKP * 4);
  int*   p1i   = (int*)ws_take(&p, (size_t)BB * NCHUNK * TOPKP * 4);
  int*   unIdx = (int*)ws_take(&p, (size_t)BB * TOPKN * 4);
  int*   ipIdx = (int*)ws_take(&p, (size_t)BB * TOPKP * 4);

  nrm_kernel<<<512, 128, 0, stream>>>(query, ctgt, qn, tn);
  mark_clear<<<(NPOOL + 255) / 256, 256, 0, stream>>>(mark);
  mark_set<<<1, 256, 0, stream>>>(indices, mark);
  tprow_kernel<<<(KK + 255) / 256, 256, 0, stream>>>(indices, index_queue,
                                                     pool_qindex, mark, tpRow);
  gemm_topk_kernel<TOPKN, 0><<<ROWG * NCHUNK, 128, 0, stream>>>(tn, queue, pool,
                                                                tpRow, p0v, p0i);
  gemm_topk_kernel<TOPKP, 1><<<ROWG * NCHUNK, 128, 0, stream>>>(tn, queue, pool,
                                                                tpRow, p1v, p1i);
  merge_kernel<TOPKN><<<1, 256, 0, stream>>>(p0v, p0i, unIdx);
  merge_kernel<TOPKP><<<1, 256, 0, stream>>>(p1v, p1i, ipIdx);
  final_kernel<<<1, 256, 0, stream>>>(tn, qn, queue, unIdx, ipIdx,
                                      labels_bank, labels, out);
}